// Memory_40089224741128
// MI455X (gfx1250) — compile-verified
//
#include <hip/hip_runtime.h>
#include <hip/hip_bf16.h>

#define MEMORY_SIZE 65536
#define KEY_SIZE    512
#define CHOOSE_K    256

typedef __attribute__((ext_vector_type(2))) float v2f;
typedef __attribute__((ext_vector_type(8))) float v8f;

// ---------------------------------------------------------------------------
// Kernel 1: scores[m] = dot(q_row0, keys[m,:]) using fp32 WMMA 16x16x4.
//
// One wave handles a 16-column tile of scores. A-matrix rows are all set to
// q[0,:] (lane L = row L, identical data per half => broadcast scalar loads),
// B-matrix column n = keys[tile*16+n, :]. Per ISA layout, lanes 0..15 carry
// K = k..k+1 and lanes 16..31 carry K = k+2..k+3 for both A and B, so the
// per-lane "half" offset is applied identically to both operands.
// D (16x16 f32): VGPR0 lanes 0..15 = row M=0, N=lane -> the score we want.
// ---------------------------------------------------------------------------
__global__ void __launch_bounds__(256)
score_wmma_kernel(const float* __restrict__ q,
                  const float* __restrict__ keys,
                  float* __restrict__ scores) {
    const int lane = threadIdx.x & 31;
    const int waveInBlock = threadIdx.x >> 5;
    const int tile = blockIdx.x * (blockDim.x >> 5) + waveInBlock; // 16 cols/tile
    const int n0   = tile * 16;
    const int col  = lane & 15;
    const int half = lane >> 4;              // 0: K+=0,1   1: K+=2,3

    const float* __restrict__ keyRow = keys + (size_t)(n0 + col) * KEY_SIZE;

    v8f c0 = {};
    v8f c1 = {};
    #pragma unroll 4
    for (int k = 0; k < KEY_SIZE; k += 8) {
        const int kk = k + half * 2;
        // A fragment: every row = q[0,:]  (uniform across lanes within a half)
        v2f a0 = *(const v2f*)(q + kk);
        v2f b0 = *(const v2f*)(keyRow + kk);
        c0 = __builtin_amdgcn_wmma_f32_16x16x4_f32(false, a0, false, b0,
                                                   (short)0, c0, false, false);
        v2f a1 = *(const v2f*)(q + kk + 4);
        v2f b1 = *(const v2f*)(keyRow + kk + 4);
        c1 = __builtin_amdgcn_wmma_f32_16x16x4_f32(false, a1, false, b1,
                                                   (short)0, c1, false, false);
    }
    // Row M=0 of D lives in VGPR0 of lanes 0..15.
    const float s = c0[0] + c1[0];
    if (half == 0) {
        scores[n0 + col] = s;
    }
}

// ---------------------------------------------------------------------------
// Kernel 2: top-256 of 65536 scores + gather of value[], single workgroup.
//
// Scores are staged into LDS (256 KB -- fits in CDNA5's 320 KB/WGP).
// Each of 1024 threads owns 64 strided elements and maintains its running
// max as a packed u64 key: (orderable_float << 32) | ~index. u64 max ==
// "larger score wins; tie -> lower index wins" (jax.lax.top_k ordering).
// Per round: tree-reduce the 1024 cached maxima (shuffle + LDS), emit the
// winner, then only the owning thread rescans its 64 LDS slots.
// ---------------------------------------------------------------------------
__device__ __forceinline__ unsigned long long pack_key(float f, int i) {
    unsigned u = __float_as_uint(f);
    u = (u & 0x80000000u) ? ~u : (u | 0x80000000u);   // total order on floats
    return ((unsigned long long)u << 32) | (unsigned)(~i);
}

__global__ void __launch_bounds__(1024)
topk256_kernel(const float* __restrict__ scores,
               const float* __restrict__ value,
               float* __restrict__ out) {
    extern __shared__ float sscore[];                     // 65536 floats (dyn)
    __shared__ unsigned long long tmax[1024];             // per-thread max key
    __shared__ unsigned long long red[32];                // per-wave partials

    const int tid  = threadIdx.x;
    const int lane = tid & 31;
    const int wave = tid >> 5;

    // Stage scores to LDS and compute initial per-thread max (coalesced).
    {
        unsigned long long m = 0ull;
        #pragma unroll 8
        for (int j = 0; j < MEMORY_SIZE / 1024; ++j) {
            const int i = tid + j * 1024;
            const float f = scores[i];
            sscore[i] = f;
            const unsigned long long k = pack_key(f, i);
            m = (k > m) ? k : m;
        }
        tmax[tid] = m;
    }
    __syncthreads();

    const float NEG_INF = -__builtin_inff();

    for (int sel = 0; sel < CHOOSE_K; ++sel) {
        // --- reduce 1024 cached maxima ---
        unsigned long long k = tmax[tid];
        #pragma unroll
        for (int off = 16; off > 0; off >>= 1) {
            const unsigned long long o = __shfl_down(k, off, 32);
            k = (o > k) ? o : k;
        }
        if (lane == 0) red[wave] = k;
        __syncthreads();
        if (wave == 0) {
            unsigned long long k2 = red[lane];
            #pragma unroll
            for (int off = 16; off > 0; off >>= 1) {
                const unsigned long long o = __shfl_down(k2, off, 32);
                k2 = (o > k2) ? o : k2;
            }
            if (lane == 0) red[0] = k2;
        }
        __syncthreads();

        const unsigned long long best = red[0];
        const int idx = (int)(~(unsigned)(best & 0xFFFFFFFFull));

        if (tid == 0) {
            out[sel] = value[idx];                         // gather in rank order
        }
        if (tid == (idx & 1023)) {                         // owner: knock out + rescan
            sscore[idx] = NEG_INF;
            unsigned long long m = 0ull;
            #pragma unroll 8
            for (int j = 0; j < MEMORY_SIZE / 1024; ++j) {
                const int i = tid + j * 1024;
                const unsigned long long kk = pack_key(sscore[i], i);
                m = (kk > m) ? kk : m;
            }
            tmax[tid] = m;
        }
        __syncthreads();
    }
}

// ---------------------------------------------------------------------------
extern "C" void kernel_launch(void* const* d_in, const int* in_sizes, int n_in,
                              void* d_out, int out_size, void* d_ws, size_t ws_size,
                              hipStream_t stream) {
    const float* q     = (const float*)d_in[0];   // [2048, 512] (only row 0 used)
    const float* keys  = (const float*)d_in[1];   // [65536, 512]
    const float* value = (const float*)d_in[2];   // [65536]
    float*       out   = (float*)d_out;           // [256]
    float*       scores = (float*)d_ws;           // 65536 floats of scratch

    // 4096 score tiles of 16 columns; 8 waves (256 threads) per block.
    score_wmma_kernel<<<MEMORY_SIZE / 16 / 8, 256, 0, stream>>>(q, keys, scores);

    // Single-WGP top-k with the whole score vector resident in LDS (256 KB).
    topk256_kernel<<<1, 1024, MEMORY_SIZE * sizeof(float), stream>>>(scores, value, out);
}